// SpatialPyramidPooling_40166534152820
// MI455X (gfx1250) — compile-verified
//
#include <hip/hip_runtime.h>
#include <hip/hip_bf16.h>

// SPP-mix (max + mean) pyramid pooling for x=(32,64,64,512) f32, levels {1,2,4}.
// Memory-bound: 256 MiB streamed once via CDNA5 async global->LDS copies
// (global_load_async_to_lds_b128 + s_wait_asynccnt), double-buffered in LDS.

#define B_   32
#define H_   64
#define W_   64
#define C_   512
#define CG   128            // C/4 channel groups (float4)
#define NBIN 16             // 4x4 level-4 bins
#define OUT_STRIDE 10752    // 21*C per batch
#define L2_OFF 512          // level-2 block starts at 1*C
#define L4_OFF 2560         // level-4 block starts at 5*C

__device__ __forceinline__ float4 f4max(float4 a, float4 b) {
    return make_float4(fmaxf(a.x,b.x), fmaxf(a.y,b.y), fmaxf(a.z,b.z), fmaxf(a.w,b.w));
}
__device__ __forceinline__ float4 f4add(float4 a, float4 b) {
    return make_float4(a.x+b.x, a.y+b.y, a.z+b.z, a.w+b.w);
}
__device__ __forceinline__ float4 f4mix(float4 m, float4 s, float inv) {
    return make_float4(m.x + s.x*inv, m.y + s.y*inv, m.z + s.z*inv, m.w + s.w*inv);
}
// Low 32 bits of the LDS-aperture flat address == LDS byte offset (ISA 10.2:
// LDS address is the low 32 bits of the flat address); used as VDST (LDS address) for async loads.
__device__ __forceinline__ unsigned lds_u32(const void* p) {
    return (unsigned)(unsigned long long)p;
}

// One block per (batch b, level-4 bin). bin = jy*4 + ix  (col-bin * 4 + row-bin),
// which is exactly the reference's flattened (jy outer, ix inner) order.
__global__ __launch_bounds__(256)
void spp_bins_kernel(const float* __restrict__ x,
                     float* __restrict__ out,
                     float4* __restrict__ ws_max,
                     float4* __restrict__ ws_sum) {
    // Two 32KB row buffers: one bin-row slice = 16 cols * 512 ch * 4B (contiguous).
    __shared__ __align__(16) float lds[2][8192];

    const int t    = threadIdx.x;            // 0..255
    const int blk  = blockIdx.x;             // 0..511
    const int b    = blk >> 4;
    const int bin  = blk & 15;               // = jy*4 + ix
    const int ix   = bin & 3;                // row-bin
    const int jy   = bin >> 2;               // col-bin
    const int r0   = ix * 16;
    const int c0   = jy * 16;
    const int g    = t & (CG - 1);           // channel group (4 channels)
    const int half = t >> 7;                 // 0: cols 0..7, 1: cols 8..15

    // Issue one bin-row (32KB contiguous) global->LDS asynchronously.
    // Per thread: 8 x B128, addresses strided so the block covers the chunk.
    auto issue_stage = [&](int s, int bufIdx) {
        const float* src = x + (((size_t)(b * H_ + (r0 + s)) * W_) + c0) * (size_t)C_;
        const unsigned lbase = lds_u32(&lds[bufIdx][0]);
#pragma unroll
        for (int k = 0; k < 8; ++k) {
            const float* gp = src + (t * 4 + k * 1024);
            unsigned     lp = lbase + (unsigned)(t * 16 + k * 4096);
            asm volatile("global_load_async_to_lds_b128 %0, %1, off"
                         :: "v"(lp), "v"(gp) : "memory");
        }
    };

    float4 vmax = make_float4(-__builtin_inff(), -__builtin_inff(),
                              -__builtin_inff(), -__builtin_inff());
    float4 vsum = make_float4(0.f, 0.f, 0.f, 0.f);

    issue_stage(0, 0);
    for (int s = 0; s < 16; ++s) {
        if (s + 1 < 16) {
            issue_stage(s + 1, (s + 1) & 1);
            // async loads complete in order: waiting to <=8 retires stage s.
            asm volatile("s_wait_asynccnt 0x8" ::: "memory");
        } else {
            asm volatile("s_wait_asynccnt 0x0" ::: "memory");
        }
        __syncthreads();   // all waves' stage-s data now in LDS

        const float4* lb = (const float4*)&lds[s & 1][0];
#pragma unroll
        for (int cc = 0; cc < 8; ++cc) {
            float4 v = lb[(half * 8 + cc) * CG + g];  // [col][ch-group]
            vmax = f4max(vmax, v);
            vsum = f4add(vsum, v);
        }
        __syncthreads();   // buffer may be overwritten by stage s+2 issue
    }

    // Combine the two column-halves through LDS (buffers are dead now).
    float4* red = (float4*)&lds[0][0];
    if (half) { red[g] = vmax; red[CG + g] = vsum; }
    __syncthreads();
    if (!half) {
        vmax = f4max(vmax, red[g]);
        vsum = f4add(vsum, red[CG + g]);
        const size_t wsi = ((size_t)(b * NBIN + bin)) * CG + g;
        ws_max[wsi] = vmax;
        ws_sum[wsi] = vsum;
        // Level-4 output: offset (jy*4+ix)*C = bin*C ; window = 16*16 = 256 elems.
        float4* o4 = (float4*)(out + (size_t)b * OUT_STRIDE + L4_OFF + bin * C_);
        o4[g] = f4mix(vmax, vsum, 1.0f / 256.0f);
    }
}

// One block per batch; thread g combines the 16 level-4 partials per channel group.
__global__ __launch_bounds__(128)
void spp_combine_kernel(const float4* __restrict__ ws_max,
                        const float4* __restrict__ ws_sum,
                        float* __restrict__ out) {
    const int b = blockIdx.x;
    const int g = threadIdx.x;

    float4 m1 = make_float4(-__builtin_inff(), -__builtin_inff(),
                            -__builtin_inff(), -__builtin_inff());
    float4 s1 = make_float4(0.f, 0.f, 0.f, 0.f);
    float4 m2[4], s2[4];
#pragma unroll
    for (int q = 0; q < 4; ++q) { m2[q] = m1; s2[q] = s1; }

#pragma unroll
    for (int bin = 0; bin < 16; ++bin) {
        const size_t wsi = ((size_t)(b * NBIN + bin)) * CG + g;
        float4 m = ws_max[wsi];
        float4 s = ws_sum[wsi];
        m1 = f4max(m1, m);
        s1 = f4add(s1, s);
        const int ix = bin & 3, jy = bin >> 2;
        const int q  = (jy >> 1) * 2 + (ix >> 1);   // level-2 (jy2 outer, ix2 inner)
        m2[q] = f4max(m2[q], m);
        s2[q] = f4add(s2[q], s);
    }

    float* ob = out + (size_t)b * OUT_STRIDE;
    ((float4*)ob)[g] = f4mix(m1, s1, 1.0f / 4096.0f);          // level 1 (64x64)
#pragma unroll
    for (int q = 0; q < 4; ++q) {                              // level 2 (32x32)
        float4* o2 = (float4*)(ob + L2_OFF + q * C_);
        o2[g] = f4mix(m2[q], s2[q], 1.0f / 1024.0f);
    }
}

extern "C" void kernel_launch(void* const* d_in, const int* in_sizes, int n_in,
                              void* d_out, int out_size, void* d_ws, size_t ws_size,
                              hipStream_t stream) {
    (void)in_sizes; (void)n_in; (void)out_size; (void)ws_size;
    const float* x   = (const float*)d_in[0];
    float*       out = (float*)d_out;
    // Workspace: per-(b,bin,channel-group) max and sum partials (1 MiB each).
    float4* ws_max = (float4*)d_ws;
    float4* ws_sum = ws_max + (size_t)B_ * NBIN * CG;

    spp_bins_kernel<<<B_ * NBIN, 256, 0, stream>>>(x, out, ws_max, ws_sum);
    spp_combine_kernel<<<B_, 128, 0, stream>>>(ws_max, ws_sum, out);
}